// ERECAPModel_28965259444770
// MI455X (gfx1250) — compile-verified
//
#include <hip/hip_runtime.h>
#include <stdint.h>

// Problem constants (fixed by the reference harness)
#define S_   8192
#define D_   2048
#define L_   8
#define H_   16
#define HD_  128
#define K_   4096

// Flat output layout (floats), concatenated in reference return order
#define N_HIDDEN   ((size_t)K_ * (size_t)D_)            // 8,388,608
#define OFF_HIDDEN ((size_t)0)
#define OFF_MASK   (N_HIDDEN)                           // 8,388,608
#define OFF_IDX    (OFF_MASK + (size_t)K_)              // 8,392,704
#define OFF_KEYS   (OFF_IDX + (size_t)K_)               // 8,396,800
#define N_KV       ((size_t)L_ * H_ * K_ * HD_)         // 67,108,864
#define OFF_VALS   (OFF_KEYS + N_KV)                    // 75,505,664

// ---------------------------------------------------------------------------
// Kernel 1: exact top-k + ascending index sort, all inside one WGP's LDS.
// 8192 (score,index) pairs = 64 KB LDS (CDNA5 has 320 KB/WGP).
// Bitonic sort: descending by score, ties -> lower index (lax.top_k order),
// then bitonic sort of the kept 4096 indices ascending (jnp.sort).
// ---------------------------------------------------------------------------
__global__ __launch_bounds__(1024) void erecap_topk_sort(
    const float* __restrict__ scores, const int* __restrict__ mask,
    float* __restrict__ out, int* __restrict__ ws_idx)
{
  __shared__ float s_key[S_];
  __shared__ int   s_val[S_];
  const int t = threadIdx.x;

  for (int i = t; i < S_; i += 1024) { s_key[i] = scores[i]; s_val[i] = i; }
  __syncthreads();

  // Phase 1: full bitonic sort, descending by (score, -index)
  for (int k = 2; k <= S_; k <<= 1) {
    for (int j = k >> 1; j > 0; j >>= 1) {
      for (int i = t; i < S_; i += 1024) {
        const int p = i ^ j;
        if (p > i) {
          const float si = s_key[i], sp = s_key[p];
          const int   vi = s_val[i], vp = s_val[p];
          const bool earlier = (si > sp) || (si == sp && vi < vp); // i first in desc order
          const bool forward = ((i & k) == 0);
          if (forward ? !earlier : earlier) {
            s_key[i] = sp; s_key[p] = si;
            s_val[i] = vp; s_val[p] = vi;
          }
        }
      }
      __syncthreads();
    }
  }

  // Phase 2: sort kept indices s_val[0..K) ascending
  for (int k = 2; k <= K_; k <<= 1) {
    for (int j = k >> 1; j > 0; j >>= 1) {
      for (int i = t; i < K_; i += 1024) {
        const int p = i ^ j;
        if (p > i) {
          const int vi = s_val[i], vp = s_val[p];
          const bool forward = ((i & k) == 0);
          if (forward ? (vi > vp) : (vi < vp)) { s_val[i] = vp; s_val[p] = vi; }
        }
      }
      __syncthreads();
    }
  }

  for (int i = t; i < K_; i += 1024) {
    const int id = s_val[i];
    ws_idx[i]           = id;             // consumed by gather kernel
    out[OFF_IDX + i]    = (float)id;      // idx output (harness compares as float)
    out[OFF_MASK + i]   = (float)mask[id];
  }
}

// ---------------------------------------------------------------------------
// Kernel 2: 512 B-chunk gather via CDNA5 async LDS data movers, 4 chunks per
// wave with a provably-safe ASYNCcnt drain ladder:
//   issue 4 loads; then wait<=3/store0, wait<=2/store1, wait<=1/store2,
//   wait<=0/store3.  ASYNCcnt mixes loads+stores and cross-type completion is
//   unordered, but loads retire in order, so after issuing (4 loads + s
//   stores), asynccnt<=c guarantees (4+s-c-s)=4-c loads complete.
// NT hints: every byte is touched exactly once -> keep 1.1 GB of streaming
// traffic from evicting live data in the 192 MB L2.
// Chunk space: [hidden: K*16] [keys: L*H*K] [values: L*H*K], all 512 B-aligned.
// ---------------------------------------------------------------------------
#define HID_CHUNKS   ((long)K_ * (D_ / HD_))        // 65,536
#define KV_CHUNKS    ((long)L_ * H_ * K_)           // 524,288
#define TOTAL_CHUNKS (HID_CHUNKS + 2 * KV_CHUNKS)   // 1,114,112 (div by 32)
#define CHUNKS_PER_WAVE 4
#define WAVES_PER_BLOCK 8

#define ASYNC_LOAD_NT(ldsaddr, gaddr)                                   \
  asm volatile("global_load_async_to_lds_b128 %0, %1, off th:TH_LOAD_NT" \
               :: "v"(ldsaddr), "v"(gaddr) : "memory")
#define ASYNC_STORE_NT(gaddr, ldsaddr)                                   \
  asm volatile("global_store_async_from_lds_b128 %0, %1, off th:TH_STORE_NT" \
               :: "v"(gaddr), "v"(ldsaddr) : "memory")

__global__ __launch_bounds__(256) void erecap_gather_async(
    const float* __restrict__ hidden, const float* __restrict__ keys,
    const float* __restrict__ values, const int* __restrict__ ws_idx,
    float* __restrict__ out)
{
  __shared__ float stage[WAVES_PER_BLOCK * CHUNKS_PER_WAVE * HD_]; // 16 KB

  const int lane = threadIdx.x & 31;
  const int wv   = threadIdx.x >> 5;
  const long chunk0 =
      ((long)blockIdx.x * WAVES_PER_BLOCK + wv) * CHUNKS_PER_WAVE;
  if (chunk0 >= TOTAL_CHUNKS) return;   // TOTAL is a multiple of 4: all-or-none

  uint64_t ga[CHUNKS_PER_WAVE], gd[CHUNKS_PER_WAVE];
#pragma unroll
  for (int q = 0; q < CHUNKS_PER_WAVE; ++q) {
    const long chunk = chunk0 + q;
    const float* src;
    float* dst;
    if (chunk < HID_CHUNKS) {
      const int k    = (int)(chunk >> 4);   // 16 chunks per hidden row
      const int part = (int)(chunk & 15);
      const int row  = ws_idx[k];
      src = hidden + (size_t)row * D_ + (size_t)part * HD_;
      dst = out + OFF_HIDDEN + (size_t)k * D_ + (size_t)part * HD_;
    } else {
      long c = chunk - HID_CHUNKS;
      const float* base = keys;
      size_t ob = OFF_KEYS;
      if (c >= KV_CHUNKS) { c -= KV_CHUNKS; base = values; ob = OFF_VALS; }
      const int k   = (int)(c % K_);
      const int lh  = (int)(c / K_);        // fused (layer, head), 0..127
      const int row = ws_idx[k];
      src = base + ((size_t)lh * S_ + (size_t)row) * HD_;
      dst = out + ob + ((size_t)lh * K_ + (size_t)k) * HD_;
    }
    ga[q] = (uint64_t)(uintptr_t)src + (uint64_t)(lane * 16);
    gd[q] = (uint64_t)(uintptr_t)dst + (uint64_t)(lane * 16);
  }

  // Per-lane 16 B LDS slots; HW adds LDS_BASE to the VGPR value.
  const uint32_t l0 =
      (uint32_t)(uintptr_t)(&stage[wv * CHUNKS_PER_WAVE * HD_]) +
      (uint32_t)(lane * 16);
  const uint32_t l1 = l0 + 512;
  const uint32_t l2 = l0 + 1024;
  const uint32_t l3 = l0 + 1536;

  // Launch all four gathers (4 x 512 B in flight per wave).
  ASYNC_LOAD_NT(l0, ga[0]);
  ASYNC_LOAD_NT(l1, ga[1]);
  ASYNC_LOAD_NT(l2, ga[2]);
  ASYNC_LOAD_NT(l3, ga[3]);

  // Drain ladder: each wait guarantees the matching load has retired.
  asm volatile("s_wait_asynccnt 0x3" ::: "memory");
  ASYNC_STORE_NT(gd[0], l0);
  asm volatile("s_wait_asynccnt 0x2" ::: "memory");
  ASYNC_STORE_NT(gd[1], l1);
  asm volatile("s_wait_asynccnt 0x1" ::: "memory");
  ASYNC_STORE_NT(gd[2], l2);
  asm volatile("s_wait_asynccnt 0x0" ::: "memory");
  ASYNC_STORE_NT(gd[3], l3);
  // outstanding stores are drained by S_ENDPGM's implicit wait-idle
}

// ---------------------------------------------------------------------------
extern "C" void kernel_launch(void* const* d_in, const int* in_sizes, int n_in,
                              void* d_out, int out_size, void* d_ws, size_t ws_size,
                              hipStream_t stream) {
  (void)in_sizes; (void)n_in; (void)out_size; (void)ws_size;
  const float* hidden = (const float*)d_in[0];
  const float* scores = (const float*)d_in[1];
  const int*   mask   = (const int*)d_in[2];
  const float* keys   = (const float*)d_in[3];
  const float* values = (const float*)d_in[4];
  // d_in[5] = keep_k, fixed at 4096 for this problem size
  float* out  = (float*)d_out;
  int* ws_idx = (int*)d_ws;   // 4096 ints of scratch

  erecap_topk_sort<<<1, 1024, 0, stream>>>(scores, mask, out, ws_idx);

  const long chunks_per_block = (long)WAVES_PER_BLOCK * CHUNKS_PER_WAVE;
  const unsigned blocks =
      (unsigned)((TOTAL_CHUNKS + chunks_per_block - 1) / chunks_per_block);
  erecap_gather_async<<<dim3(blocks), 256, 0, stream>>>(hidden, keys, values,
                                                        ws_idx, out);
}